// SAAttention_55594056679639
// MI455X (gfx1250) — compile-verified
//
#include <hip/hip_runtime.h>
#include <math.h>

#define B_ 2
#define I_ 256
#define D_ 768
#define H_ 12
#define DP_ 128
#define DK_ 64
#define NPQ_ 4
#define NPV_ 8

typedef float v2f __attribute__((ext_vector_type(2)));
typedef float v8f __attribute__((ext_vector_type(8)));

// ---------------------------------------------------------------------------
// Generic batched GEMM on the WMMA f32 16x16x4 pipe.
//   C[by,bz][m,n] = alpha * sum_k A[m,k]*B[k,n] (+ bias[n])
// A row-major (lda); B row-major (ldb) if OPB==0 else B[n*ldb+k]; C row-major.
// Per-batch base offsets via sAy/sAz etc. (blockIdx.y / blockIdx.z).
// GUARD handles M%16 / N%16 edges branch-free: clamp address + 0/1 mask.
// K must be a multiple of 4 (true for all launches: 64/128/256/768/2304).
// ---------------------------------------------------------------------------
struct GemmP {
  const float* A; const float* B; float* C; const float* bias;
  int M, N, K;
  int lda, ldb, ldc;
  long sAy, sAz, sBy, sBz, sCy, sCz;
  float alpha;
};

template <int OPB, bool GUARD>
__global__ void ipa_gemm_wmma(GemmP p) {
  const int tilesN = (p.N + 15) >> 4;
  const int tilesM = (p.M + 15) >> 4;
  const int wave = (int)((blockIdx.x * blockDim.x + threadIdx.x) >> 5);
  if (wave >= tilesM * tilesN) return;          // uniform per wave
  const int lane = threadIdx.x & 31;
  const int tm = wave / tilesN, tn = wave % tilesN;
  const int row0 = tm << 4, col0 = tn << 4;

  const float* A = p.A + (long)blockIdx.y * p.sAy + (long)blockIdx.z * p.sAz;
  const float* B = p.B + (long)blockIdx.y * p.sBy + (long)blockIdx.z * p.sBz;
  float*       C = p.C + (long)blockIdx.y * p.sCy + (long)blockIdx.z * p.sCz;

  const int mn = lane & 15;          // M index for A frag, N index for B/C frags
  const int kb = (lane >> 4) << 1;   // K sub-offset: lanes 0-15 -> {0,1}, 16-31 -> {2,3}

  int aRow = row0 + mn;
  int bCol = col0 + mn;
  float aMask = 1.f, bMask = 1.f;
  if (GUARD) {
    if (aRow >= p.M) { aRow = p.M - 1; aMask = 0.f; }   // clamp: stay in-bounds
    if (bCol >= p.N) { bCol = p.N - 1; bMask = 0.f; }
  }
  const float* aPtr = A + (long)aRow * p.lda + kb;
  const float* bPtr = (OPB == 0) ? (B + (long)kb * p.ldb + bCol)
                                 : (B + (long)bCol * p.ldb + kb);

  v8f acc = {0.f, 0.f, 0.f, 0.f, 0.f, 0.f, 0.f, 0.f};
#pragma unroll 4
  for (int k0 = 0; k0 < p.K; k0 += 4) {
    v2f a, b;
    a.x = aPtr[k0];
    a.y = aPtr[k0 + 1];
    if (OPB == 0) {
      const float* bp = bPtr + (long)k0 * p.ldb;
      b.x = bp[0];
      b.y = bp[p.ldb];
    } else {
      b.x = bPtr[k0];
      b.y = bPtr[k0 + 1];
    }
    if (GUARD) {
      a.x *= aMask; a.y *= aMask;
      b.x *= bMask; b.y *= bMask;
    }
    acc = __builtin_amdgcn_wmma_f32_16x16x4_f32(false, a, false, b,
                                                (short)0, acc, false, false);
  }

  // C/D layout: VGPR r holds row (r + 8*lane[4]), col = lane&15
  const int cCol = col0 + mn;
  const int mo = (lane >> 4) << 3;
  const bool cColOK = GUARD ? (cCol < p.N) : true;
  if (cColOK) {
    for (int r = 0; r < 8; ++r) {
      const int cRow = row0 + mo + r;
      if (!GUARD || cRow < p.M) {
        float val = acc[r] * p.alpha;
        if (p.bias) val += p.bias[cCol];
        C[(long)cRow * p.ldc + cCol] = val;
      }
    }
  }
}

// ---------------------------------------------------------------------------
// In-place rigid-frame transform: pts[b,i,h,c,:] = R^T(b,i) @ pts + T(b,i)
// (reference: affine(pts,T,IR) with IR = swapaxes(R,-1,-2))
// ---------------------------------------------------------------------------
__global__ void ipa_affine(float* pts, const float* T, const float* R, int np, int total) {
  int idx = blockIdx.x * blockDim.x + threadIdx.x;
  if (idx >= total) return;
  int bi = idx / (H_ * np);
  const float* Rm = R + (long)bi * 9;
  const float* Tv = T + (long)bi * 3;
  float x = pts[idx * 3 + 0], y = pts[idx * 3 + 1], z = pts[idx * 3 + 2];
  float o0 = Rm[0] * x + Rm[3] * y + Rm[6] * z + Tv[0];  // sum_q R[q,p]*pt[q]
  float o1 = Rm[1] * x + Rm[4] * y + Rm[7] * z + Tv[1];
  float o2 = Rm[2] * x + Rm[5] * y + Rm[8] * z + Tv[2];
  pts[idx * 3 + 0] = o0; pts[idx * 3 + 1] = o1; pts[idx * 3 + 2] = o2;
}

// ---------------------------------------------------------------------------
// logits[b,h,i,j] += point_past + point_last + pair_bias_proj + bias
// ---------------------------------------------------------------------------
__global__ void ipa_combine(float* logits,
                            const float* qgp, const float* kgp,
                            const float* qgl, const float* kgl,
                            const float* pairproj, const float* bias,
                            const float* tpw_past, const float* tpw_last) {
  long idx = (long)blockIdx.x * blockDim.x + threadIdx.x;
  const long total = (long)B_ * H_ * I_ * I_;
  if (idx >= total) return;
  int j = (int)(idx & (I_ - 1));
  long t = idx >> 8;
  int i = (int)(t & (I_ - 1));
  t >>= 8;
  int h = (int)(t % H_);
  int b = (int)(t / H_);

  const float point_w = 0.13608276348795434f;  // 1/sqrt(3*4*9/2)
  float pwp = -0.5f * point_w * log1pf(expf(tpw_past[h]));
  float pwl = -0.5f * point_w * log1pf(expf(tpw_last[h]));

  const float* qp = qgp + ((long)(b * I_ + i) * H_ + h) * (NPQ_ * 3);
  const float* kp = kgp + ((long)(b * I_ + j) * H_ + h) * (NPQ_ * 3);
  const float* ql = qgl + ((long)(b * I_ + i) * H_ + h) * (NPQ_ * 3);
  const float* kl = kgl + ((long)(b * I_ + j) * H_ + h) * (NPQ_ * 3);

  float sp = 0.f, sl = 0.f;
  for (int c = 0; c < NPQ_; ++c) {
    float dx = qp[c * 3 + 0] - kp[c * 3 + 0];
    float dy = qp[c * 3 + 1] - kp[c * 3 + 1];
    float dz = qp[c * 3 + 2] - kp[c * 3 + 2];
    sp += sqrtf(dx * dx + dy * dy + dz * dz);
    dx = ql[c * 3 + 0] - kl[c * 3 + 0];
    dy = ql[c * 3 + 1] - kl[c * 3 + 1];
    dz = ql[c * 3 + 2] - kl[c * 3 + 2];
    sl += sqrtf(dx * dx + dy * dy + dz * dz);
  }
  float pv = pairproj[((long)(b * I_ + i) * I_ + j) * H_ + h];  // pair_w pre-folded
  logits[idx] += pwp * sp + pwl * sl + pv + bias[idx];
}

// ---------------------------------------------------------------------------
// Row softmax over j (row length I_=256), one block per (b,h,i).
// ---------------------------------------------------------------------------
__global__ void ipa_softmax(float* logits) {
  __shared__ float red[I_];
  float* row = logits + (long)blockIdx.x * I_;
  int t = threadIdx.x;
  float v = row[t];
  red[t] = v;
  __syncthreads();
  for (int s = I_ / 2; s > 0; s >>= 1) {
    if (t < s) red[t] = fmaxf(red[t], red[t + s]);
    __syncthreads();
  }
  float mx = red[0];
  __syncthreads();
  float e = expf(v - mx);
  red[t] = e;
  __syncthreads();
  for (int s = I_ / 2; s > 0; s >>= 1) {
    if (t < s) red[t] += red[t + s];
    __syncthreads();
  }
  row[t] = e / red[0];
}

// ---------------------------------------------------------------------------
// Inverse affine (always with PAST pose, faithful to reference) + norms,
// written directly into feat columns.
// ---------------------------------------------------------------------------
__global__ void ipa_inv_affine(const float* og, const float* T, const float* R,
                               float* feat, int colOl, int colN, int total) {
  int idx = blockIdx.x * blockDim.x + threadIdx.x;
  if (idx >= total) return;
  int c = idx & (NPV_ - 1);
  int t = idx >> 3;
  int h = t % H_;
  int bi = t / H_;
  const float* o = og + (long)idx * 3;
  const float* Tv = T + (long)bi * 3;
  const float* Rm = R + (long)bi * 9;
  float x = o[0] - Tv[0], y = o[1] - Tv[1], z = o[2] - Tv[2];
  float p0 = Rm[0] * x + Rm[1] * y + Rm[2] * z;  // ol[p] = sum_q R[p,q]*X[q]
  float p1 = Rm[3] * x + Rm[4] * y + Rm[5] * z;
  float p2 = Rm[6] * x + Rm[7] * y + Rm[8] * z;
  float* fr = feat + (long)bi * (2 * D_ + 2 * H_ * NPV_ * 4);
  int tc = h * (NPV_ * 3) + c * 3;
  fr[colOl + tc + 0] = p0;
  fr[colOl + tc + 1] = p1;
  fr[colOl + tc + 2] = p2;
  fr[colN + h * NPV_ + c] = sqrtf(p0 * p0 + p1 * p1 + p2 * p2);
}

// ---------------------------------------------------------------------------
extern "C" void kernel_launch(void* const* d_in, const int* in_sizes, int n_in,
                              void* d_out, int out_size, void* d_ws, size_t ws_size,
                              hipStream_t stream) {
  (void)in_sizes; (void)n_in; (void)out_size; (void)ws_size;
  const float* x1d      = (const float*)d_in[0];
  const float* x2d      = (const float*)d_in[1];
  const float* past_T   = (const float*)d_in[2];
  const float* past_R   = (const float*)d_in[3];
  const float* last_T   = (const float*)d_in[4];
  const float* last_R   = (const float*)d_in[5];
  const float* bias     = (const float*)d_in[6];
  const float* Wq       = (const float*)d_in[7];
  const float* Wk       = (const float*)d_in[8];
  const float* Wv       = (const float*)d_in[9];
  const float* Wpb      = (const float*)d_in[10];
  const float* Wpq_past = (const float*)d_in[11];
  const float* Wpk_past = (const float*)d_in[12];
  const float* Wpv_past = (const float*)d_in[13];
  const float* Wpq_last = (const float*)d_in[14];
  const float* Wpk_last = (const float*)d_in[15];
  const float* Wpv_last = (const float*)d_in[16];
  const float* tpw_past = (const float*)d_in[17];
  const float* tpw_last = (const float*)d_in[18];
  const float* Wpv      = (const float*)d_in[19];
  const float* W_out    = (const float*)d_in[20];
  const float* b_out    = (const float*)d_in[21];
  float* out = (float*)d_out;
  float* ws  = (float*)d_ws;

  const int M  = B_ * I_;                     // 512
  const int FE = 2 * D_ + 2 * H_ * NPV_ * 4;  // 2304

  // ---- workspace layout (floats); total ~7.18M floats (~28.7 MB) ----
  float* q    = ws;                   // [B,I,H,DK]        393216
  float* k    = q + 393216;
  float* v    = k + 393216;
  float* qgp  = v + 393216;           // [B,I,H,NPQ,3]      73728
  float* kgp  = qgp + 73728;
  float* vgp  = kgp + 73728;          // [B,I,H,NPV,3]     147456
  float* qgl  = vgp + 147456;
  float* kgl  = qgl + 73728;
  float* vgl  = kgl + 73728;
  float* logits   = vgl + 147456;     // [B,H,I,I]        1572864
  float* pairproj = logits + 1572864; // [(b,i,j),H]      1572864
  float* ogp  = pairproj + 1572864;   // [B,I,H,NPV,3]     147456
  float* ogl  = ogp + 147456;
  float* z    = ogl + 147456;         // [B,I,H,DP]        786432
  float* feat = z + 786432;           // [B*I, 2304]      1179648

  auto gemm = [&](const float* A, const float* Bm, float* C,
                  int Mm, int Nn, int Kk, int lda, int ldb, int ldc,
                  long sAy, long sAz, long sBy, long sBz, long sCy, long sCz,
                  int gy, int gz, int opB, float alpha, const float* bia) {
    GemmP p{A, Bm, C, bia, Mm, Nn, Kk, lda, ldb, ldc,
            sAy, sAz, sBy, sBz, sCy, sCz, alpha};
    int waves = ((Mm + 15) / 16) * ((Nn + 15) / 16);
    dim3 grid((unsigned)((waves + 7) / 8), (unsigned)gy, (unsigned)gz);
    dim3 blk(256);
    bool guard = (Mm & 15) || (Nn & 15);
    if (opB == 0) {
      if (guard) ipa_gemm_wmma<0, true ><<<grid, blk, 0, stream>>>(p);
      else       ipa_gemm_wmma<0, false><<<grid, blk, 0, stream>>>(p);
    } else {
      if (guard) ipa_gemm_wmma<1, true ><<<grid, blk, 0, stream>>>(p);
      else       ipa_gemm_wmma<1, false><<<grid, blk, 0, stream>>>(p);
    }
  };

  // ---- 1. scalar + point projections: X[512,768] @ W ----
  gemm(x1d, Wq, q, M, D_, D_, D_, D_, D_, 0,0,0,0,0,0, 1,1, 0, 1.f, nullptr);
  gemm(x1d, Wk, k, M, D_, D_, D_, D_, D_, 0,0,0,0,0,0, 1,1, 0, 1.f, nullptr);
  gemm(x1d, Wv, v, M, D_, D_, D_, D_, D_, 0,0,0,0,0,0, 1,1, 0, 1.f, nullptr);
  gemm(x1d, Wpq_past, qgp, M, 144, D_, D_, 144, 144, 0,0,0,0,0,0, 1,1, 0, 1.f, nullptr);
  gemm(x1d, Wpk_past, kgp, M, 144, D_, D_, 144, 144, 0,0,0,0,0,0, 1,1, 0, 1.f, nullptr);
  gemm(x1d, Wpv_past, vgp, M, 288, D_, D_, 288, 288, 0,0,0,0,0,0, 1,1, 0, 1.f, nullptr);
  gemm(x1d, Wpq_last, qgl, M, 144, D_, D_, 144, 144, 0,0,0,0,0,0, 1,1, 0, 1.f, nullptr);
  gemm(x1d, Wpk_last, kgl, M, 144, D_, D_, 144, 144, 0,0,0,0,0,0, 1,1, 0, 1.f, nullptr);
  gemm(x1d, Wpv_last, vgl, M, 288, D_, D_, 288, 288, 0,0,0,0,0,0, 1,1, 0, 1.f, nullptr);

  // ---- 2. rigid-frame transforms (in place) ----
  {
    int t4 = B_ * I_ * H_ * NPQ_, t8 = B_ * I_ * H_ * NPV_;
    ipa_affine<<<(t4 + 255) / 256, 256, 0, stream>>>(qgp, past_T, past_R, NPQ_, t4);
    ipa_affine<<<(t4 + 255) / 256, 256, 0, stream>>>(kgp, past_T, past_R, NPQ_, t4);
    ipa_affine<<<(t8 + 255) / 256, 256, 0, stream>>>(vgp, past_T, past_R, NPV_, t8);
    ipa_affine<<<(t4 + 255) / 256, 256, 0, stream>>>(qgl, last_T, last_R, NPQ_, t4);
    ipa_affine<<<(t4 + 255) / 256, 256, 0, stream>>>(kgl, last_T, last_R, NPQ_, t4);
    ipa_affine<<<(t8 + 255) / 256, 256, 0, stream>>>(vgl, last_T, last_R, NPV_, t8);
  }

  // ---- 3. scalar attention logits: per (h,b), Q[I,64] @ K^T -> logits[b,h,i,j]
  const float scalar_w = 0.07216878364870323f;  // 1/sqrt(3*64)
  gemm(q, k, logits, I_, I_, DK_, D_, D_, I_,
       64, (long)I_ * D_, 64, (long)I_ * D_, (long)I_ * I_, (long)H_ * I_ * I_,
       H_, B_, 1, scalar_w, nullptr);

  // ---- 4. pair bias projection: x2d[131072,128] @ Wpb[128,12] (pair_w folded) ----
  const float pair_w = 0.5773502691896258f;     // 1/sqrt(3)
  gemm(x2d, Wpb, pairproj, B_ * I_ * I_, H_, DP_, DP_, H_, H_,
       0,0,0,0,0,0, 1,1, 0, pair_w, nullptr);

  // ---- 5. combine point terms + pair + bias, then softmax ----
  {
    long total = (long)B_ * H_ * I_ * I_;
    ipa_combine<<<(unsigned)((total + 255) / 256), 256, 0, stream>>>(
        logits, qgp, kgp, qgl, kgl, pairproj, bias, tpw_past, tpw_last);
    ipa_softmax<<<B_ * H_ * I_, I_, 0, stream>>>(logits);
  }

  // ---- 6. out_scalar = attn @ v -> feat[:,768:1536] ----
  gemm(logits, v, feat + D_, I_, DK_, I_, I_, D_, FE,
       (long)I_ * I_, (long)H_ * I_ * I_, 64, (long)I_ * D_, 64, (long)I_ * FE,
       H_, B_, 0, 1.f, nullptr);

  // ---- 7. og = attn @ vg (both branches), N=24 (guarded) ----
  gemm(logits, vgp, ogp, I_, NPV_ * 3, I_, I_, H_ * NPV_ * 3, H_ * NPV_ * 3,
       (long)I_ * I_, (long)H_ * I_ * I_, NPV_ * 3, (long)I_ * H_ * NPV_ * 3,
       NPV_ * 3, (long)I_ * H_ * NPV_ * 3, H_, B_, 0, 1.f, nullptr);
  gemm(logits, vgl, ogl, I_, NPV_ * 3, I_, I_, H_ * NPV_ * 3, H_ * NPV_ * 3,
       (long)I_ * I_, (long)H_ * I_ * I_, NPV_ * 3, (long)I_ * H_ * NPV_ * 3,
       NPV_ * 3, (long)I_ * H_ * NPV_ * 3, H_, B_, 0, 1.f, nullptr);

  // ---- 8. reassociated pair value: z[b,i,h,p] = sum_j attn[b,h,i,j]*x2d[b,i,j,p]
  //      per (i,b): A[h,j] (M=12 guarded) @ B[j,p]
  gemm(logits, x2d, z, H_, DP_, I_,
       I_ * I_, DP_, DP_,
       (long)I_, (long)H_ * I_ * I_,          // A: +i*256, +b*H*I*I
       (long)I_ * DP_, (long)I_ * I_ * DP_,   // B: +i*I*128, +b*I*I*128
       (long)H_ * DP_, (long)I_ * H_ * DP_,   // C: +i*H*128, +b*I*H*128
       I_, B_, 0, 1.f, nullptr);

  // ---- 9. out_pair = z @ W_pair_value (per h) -> feat[:,0:768] ----
  gemm(z, Wpv, feat, I_, DK_, DP_, H_ * DP_, D_, FE,
       DP_, (long)I_ * H_ * DP_, 64, 0, 64, (long)I_ * FE,
       H_, B_, 0, 1.f, nullptr);

  // ---- 10. inverse affine (PAST pose for both, faithful) + norms -> feat ----
  {
    int t8 = B_ * I_ * H_ * NPV_;
    ipa_inv_affine<<<(t8 + 255) / 256, 256, 0, stream>>>(
        ogp, past_T, past_R, feat, 2 * D_, 2 * D_ + 288, t8);
    ipa_inv_affine<<<(t8 + 255) / 256, 256, 0, stream>>>(
        ogl, past_T, past_R, feat, 2 * D_ + 384, 2 * D_ + 384 + 288, t8);
  }

  // ---- 11. final: feat[512,2304] @ W_out[2304,768] + b_out ----
  gemm(feat, W_out, out, M, D_, FE, FE, D_, D_,
       0,0,0,0,0,0, 1,1, 0, 1.f, b_out);
}